// Pol2VecBinary_4870492914036
// MI455X (gfx1250) — compile-verified
//
#include <hip/hip_runtime.h>
#include <math.h>

typedef __attribute__((ext_vector_type(2))) float v2f;
typedef __attribute__((ext_vector_type(8))) float v8f;

#define VORDER 8      // VAR_SIZE = ORDER+1
#define NROW   2048
#define NBATCH 2048
#define DIM    64
#define KPAD   576    // 512 (o,d) + 36 Gram pairs + 1 norm col + pad
#define KNORM  548    // 512 + 36

// ---------------------------------------------------------------------------
// Kernel A: build W[b, k] (the per-batch GEMM operand) + gather gamma_cols.
//   W[b, o*64+d] = -2 * T[o,b] * c_b[d]
//   W[b, 512+p]  = mult * T[i,b]*T[j,b]   (p enumerates i<=j, mult=2 if i!=j)
//   W[b, 548]    = ||c_b||^2 ; rest zero
// ---------------------------------------------------------------------------
__global__ void build_w_kernel(const int* __restrict__ col_indices,
                               const float* __restrict__ col_times,
                               const float* __restrict__ z_cols,
                               const float* __restrict__ gamma_cols,
                               float* __restrict__ W,
                               float* __restrict__ gcol) {
  int b = blockIdx.x * blockDim.x + threadIdx.x;
  if (b >= NBATCH) return;
  float t = col_times[b];
  float T[VORDER];
  T[0] = 1.0f;
  #pragma unroll
  for (int o = 1; o < VORDER; ++o) T[o] = T[o - 1] * t / (float)o;  // t^o/o!
  int idx = col_indices[b];
  const float* c = z_cols + (size_t)idx * DIM;
  float* w = W + (size_t)b * KPAD;
  float nrm = 0.0f;
  for (int d = 0; d < DIM; ++d) {
    float cd = c[d];
    nrm += cd * cd;
    #pragma unroll
    for (int o = 0; o < VORDER; ++o) w[o * DIM + d] = -2.0f * T[o] * cd;
  }
  int p = 0;
  #pragma unroll
  for (int i = 0; i < VORDER; ++i)
    #pragma unroll
    for (int j = i; j < VORDER; ++j) {
      float mult = (i == j) ? 1.0f : 2.0f;
      w[512 + p] = mult * T[i] * T[j];
      ++p;
    }
  w[KNORM] = nrm;
  for (int k = KNORM + 1; k < KPAD; ++k) w[k] = 0.0f;
  gcol[b] = gamma_cols[idx];
}

// ---------------------------------------------------------------------------
// Kernel B: build A_N[n, k] (the per-row GEMM operand). One wave per row n;
// each lane owns 2 of the 64 dims. Gram entries via wave32 shuffle reduction.
//   A_N[n, o*64+d] = z_rows[o,n,d]
//   A_N[n, 512+p]  = sum_d z_rows[i,n,d]*z_rows[j,n,d]
//   A_N[n, 548]    = 1 ; rest zero
// ---------------------------------------------------------------------------
__global__ void build_a_kernel(const float* __restrict__ z_rows,
                               float* __restrict__ A_N) {
  int gwave = (blockIdx.x * blockDim.x + threadIdx.x) >> 5;
  int lane  = threadIdx.x & 31;
  if (gwave >= NROW) return;
  int n = gwave;
  float z[VORDER][2];
  #pragma unroll
  for (int o = 0; o < VORDER; ++o) {
    const float* src = z_rows + ((size_t)o * NROW + n) * DIM + lane * 2;
    v2f v = *(const v2f*)src;
    z[o][0] = v.x; z[o][1] = v.y;
    *(v2f*)(A_N + (size_t)n * KPAD + o * DIM + lane * 2) = v;
  }
  int p = 0;
  #pragma unroll
  for (int i = 0; i < VORDER; ++i)
    #pragma unroll
    for (int j = i; j < VORDER; ++j) {
      float v = z[i][0] * z[j][0] + z[i][1] * z[j][1];
      #pragma unroll
      for (int off = 16; off > 0; off >>= 1) v += __shfl_down(v, off, 32);
      if (lane == 0) A_N[(size_t)n * KPAD + 512 + p] = v;
      ++p;
    }
  if (lane == 0) {
    A_N[(size_t)n * KPAD + KNORM] = 1.0f;
    for (int k = KNORM + 1; k < KPAD; ++k) A_N[(size_t)n * KPAD + k] = 0.0f;
  }
}

// ---------------------------------------------------------------------------
// Kernel C: fused GEMM (f32 WMMA 16x16x4, K=576) + NLL epilogue + reduction.
// Block = 256 threads = 8 waves; each wave owns a 16(batch) x 16(row) tile,
// block tile = 32 x 64. dist2 comes straight out of the accumulator.
// ---------------------------------------------------------------------------
__global__ void __launch_bounds__(256)
gemm_loss_kernel(const float* __restrict__ W,
                 const float* __restrict__ A_N,
                 const float* __restrict__ gcol,
                 const float* __restrict__ gamma_rows,
                 const int* __restrict__ events,
                 float* __restrict__ out) {
  int tid  = threadIdx.x;
  int lane = tid & 31;
  int wv   = tid >> 5;        // 0..7
  int wm   = wv >> 2;         // 0..1  (batch-tile within block)
  int wn   = wv & 3;          // 0..3  (row-tile within block)
  int m0 = blockIdx.x * 32 + wm * 16;   // batch origin
  int n0 = blockIdx.y * 64 + wn * 16;   // row origin
  int lh  = lane & 15;
  int khi = (lane >> 4) << 1;           // lanes 16-31 hold K+2,K+3

  const float* wrow = W   + (size_t)(m0 + lh) * KPAD + khi;
  const float* arow = A_N + (size_t)(n0 + lh) * KPAD + khi;

  v8f acc = {};
  #pragma unroll 4
  for (int k = 0; k < KPAD; k += 4) {
    v2f a = *(const v2f*)(wrow + k);   // A operand: W[m, k..k+3]
    v2f b = *(const v2f*)(arow + k);   // B operand: A_N[n, k..k+3]
    acc = __builtin_amdgcn_wmma_f32_16x16x4_f32(
        /*neg_a=*/false, a, /*neg_b=*/false, b,
        /*c_mod=*/(short)0, acc, /*reuse_a=*/false, /*reuse_b=*/false);
  }

  // Epilogue: acc[g] = dist2 for (b = m0 + M, n = n0 + N) with
  // M = g + 8*(lane>=16), N = lane&15.
  float lsum = 0.0f;
  int mbase = m0 + ((lane >> 4) << 3);
  int nidx  = n0 + lh;
  float gr = gamma_rows[nidx];
  #pragma unroll
  for (int g = 0; g < 8; ++g) {
    int bidx = mbase + g;
    float dist = sqrtf(fmaxf(acc[g], 0.0f));
    float x = gr + gcol[bidx] - dist;                 // logit
    int e = events[(size_t)nidx * NBATCH + bidx];
    float y = e ? -x : x;                             // -log p_mat = softplus(y)
    lsum += fmaxf(y, 0.0f) + log1pf(expf(-fabsf(y)));
  }
  #pragma unroll
  for (int off = 16; off > 0; off >>= 1) lsum += __shfl_down(lsum, off, 32);
  if (lane == 0) atomicAdd(out, lsum);
}

__global__ void zero_out_kernel(float* out) {
  if (blockIdx.x == 0 && threadIdx.x == 0) out[0] = 0.0f;
}

extern "C" void kernel_launch(void* const* d_in, const int* in_sizes, int n_in,
                              void* d_out, int out_size, void* d_ws, size_t ws_size,
                              hipStream_t stream) {
  const int*   events      = (const int*)d_in[0];    // [2048, 2048] (n, b)
  const int*   col_indices = (const int*)d_in[1];    // [2048]
  const float* col_times   = (const float*)d_in[2];  // [2048]
  const float* z_rows      = (const float*)d_in[3];  // [8, 2048, 64]
  const float* z_cols      = (const float*)d_in[4];  // [16384, 64]
  const float* gamma_rows  = (const float*)d_in[5];  // [2048]
  const float* gamma_cols  = (const float*)d_in[6];  // [16384]
  float* out = (float*)d_out;

  float* W    = (float*)d_ws;                        // [2048, 576]
  float* A_N  = W + (size_t)NBATCH * KPAD;           // [2048, 576]
  float* gcol = A_N + (size_t)NROW * KPAD;           // [2048]

  zero_out_kernel<<<1, 64, 0, stream>>>(out);
  build_w_kernel<<<NBATCH / 256, 256, 0, stream>>>(col_indices, col_times,
                                                   z_cols, gamma_cols, W, gcol);
  build_a_kernel<<<NROW / 8, 256, 0, stream>>>(z_rows, A_N);
  dim3 grid(NBATCH / 32, NROW / 64);
  gemm_loss_kernel<<<grid, 256, 0, stream>>>(W, A_N, gcol, gamma_rows, events, out);
}